// EvolveGCNO_87926570484611
// MI455X (gfx1250) — compile-verified
//
#include <hip/hip_runtime.h>
#include <math.h>

#define T_SNAP 4
#define NNODE  20000
#define NEDGE  320000
#define FIN    166
#define HID    200
#define CHH    307
#define NCC    100
#define SLOPE  (11.0f/48.0f)

typedef float v2f __attribute__((ext_vector_type(2)));
typedef float v8f __attribute__((ext_vector_type(8)));

// ---------------- utility ----------------
__global__ void fill_kernel(float* __restrict__ p, float v, long n) {
  long i = (long)blockIdx.x * blockDim.x + threadIdx.x;
  if (i < n) p[i] = v;
}

// degree accumulation: ns[t,src]+=1, nd[t,dst]+=1
__global__ void deg_kernel(const int* __restrict__ src, const int* __restrict__ dst,
                           float* __restrict__ ns, float* __restrict__ nd) {
  long i = (long)blockIdx.x * blockDim.x + threadIdx.x;
  long total = (long)T_SNAP * NEDGE;
  if (i >= total) return;
  int t = (int)(i / NEDGE);
  atomicAdd(&ns[(long)t * NNODE + src[i]], 1.0f);
  atomicAdd(&nd[(long)t * NNODE + dst[i]], 1.0f);
}

// deg -> norm factor: v>0 ? rsqrt(max(v,1)) : 0
__global__ void norm_kernel(float* __restrict__ p, long n) {
  long i = (long)blockIdx.x * blockDim.x + threadIdx.x;
  if (i >= n) return;
  float v = p[i];
  p[i] = (v > 0.f) ? rsqrtf(fmaxf(v, 1.f)) : 0.f;
}

// ---------------- Matrix-GRU weight evolution (one workgroup) ----------------
__global__ void gru_kernel(const float* __restrict__ uW, const float* __restrict__ uU,
                           const float* __restrict__ ub,
                           const float* __restrict__ rW, const float* __restrict__ rU,
                           const float* __restrict__ rb,
                           const float* __restrict__ hW, const float* __restrict__ hU,
                           const float* __restrict__ hb,
                           const float* __restrict__ W0, float* __restrict__ Ws,
                           float* __restrict__ Q, float* __restrict__ u,
                           float* __restrict__ rr, float* __restrict__ hc,
                           int r, int H) {
  int tid = threadIdx.x, nt = blockDim.x;
  int sz = r * H;
  for (int idx = tid; idx < sz; idx += nt) Q[idx] = W0[idx];
  __syncthreads();
  for (int t = 0; t < T_SNAP; ++t) {
    // phase A: u = sigmoid((uW+uU)@Q + ub); rr = sigmoid((rW+rU)@Q + rb) * Q
    for (int idx = tid; idx < sz; idx += nt) {
      int i = idx / H, j = idx - i * H;
      float su = ub[idx], sr = rb[idx];
      for (int k = 0; k < r; ++k) {
        float q = Q[k * H + j];
        su += (uW[i * r + k] + uU[i * r + k]) * q;
        sr += (rW[i * r + k] + rU[i * r + k]) * q;
      }
      u[idx]  = 1.f / (1.f + expf(-su));
      rr[idx] = (1.f / (1.f + expf(-sr))) * Q[idx];
    }
    __syncthreads();
    // phase B: hc = tanh(hW@Q + hU@rr + hb)
    for (int idx = tid; idx < sz; idx += nt) {
      int i = idx / H, j = idx - i * H;
      float sh = hb[idx];
      for (int k = 0; k < r; ++k)
        sh += hW[i * r + k] * Q[k * H + j] + hU[i * r + k] * rr[k * H + j];
      hc[idx] = tanhf(sh);
    }
    __syncthreads();
    // phase C: Q = (1-u)*Q + u*hc ; Ws[t] = Q
    for (int idx = tid; idx < sz; idx += nt) {
      float un = u[idx];
      float nQ = (1.f - un) * Q[idx] + un * hc[idx];
      Q[idx] = nQ;
      Ws[(long)t * sz + idx] = nQ;
    }
    __syncthreads();
  }
}

// ---------------- scatter aggregate: agg[t,dst,:] += x[t,src,:]*ns[t,src] ----------------
__global__ void scatter_kernel(const float* __restrict__ x, const int* __restrict__ src,
                               const int* __restrict__ dst, const float* __restrict__ ns,
                               float* __restrict__ agg, int F) {
  int t = blockIdx.y;
  int e = blockIdx.x * blockDim.y + threadIdx.y;
  if (e >= NEDGE) return;
  int s = src[(long)t * NEDGE + e];
  int d = dst[(long)t * NEDGE + e];
  float sc = ns[(long)t * NNODE + s];
  const float* xp = x + ((long)t * NNODE + s) * F;
  float* ap = agg + ((long)t * NNODE + d) * F;
  for (int f = threadIdx.x; f < F; f += blockDim.x)
    atomicAdd(&ap[f], xp[f] * sc);
}

// ---------------- WMMA fp32 GEMM: out[t] = act( (rowscale*A[t]) @ W[t] + bias ) ----------------
// One wave computes a 32x32 output tile = 4 accumulators (2 row-blocks x 2 col-blocks)
// using V_WMMA_F32_16X16X4_F32, K stepped by 4. Branch-free inner loop:
//  - main loop covers K & ~3 (no guards at all)
//  - B column indices are clamped (hoisted) — garbage columns never stored
//  - single masked epilogue step for K % 4 (A-side zero-mask kills the product)
// A fragment (ISA 7.12.2): lanes 0-15 hold K=k,k+1; lanes 16-31 hold K=k+2,k+3.
// act: 0=none, 1=RReLU(eval), 2=ReLU
#define WMMA4(a, b, c) __builtin_amdgcn_wmma_f32_16x16x4_f32(false, a, false, b, (short)0, c, false, false)

__global__ void wmma_gemm_kernel(const float* __restrict__ A,
                                 const float* __restrict__ W, long wStride,
                                 const float* __restrict__ bias,
                                 const float* __restrict__ rowscale,
                                 float* __restrict__ out,
                                 int M, int K, int Nout, int act) {
  int t    = blockIdx.y;
  int lane = threadIdx.x & 31;
  int wave = threadIdx.x >> 5;
  int tilesN   = (Nout + 31) >> 5;
  int totTiles = (M >> 5) * tilesN;
  int tile = blockIdx.x * 4 + wave;
  if (tile >= totTiles) return;
  int tm = tile / tilesN, tn = tile - tm * tilesN;
  int m0 = tm << 5, n0 = tn << 5;
  int half = lane >> 4, l = lane & 15;

  const float* Ab = A + (long)t * M * K;
  const float* Wb = W + (long)t * wStride;

  int row0 = m0 + l;
  int row1 = m0 + 16 + l;
  int col0 = n0 + l;
  int col1 = n0 + 16 + l;
  int colm0 = (col0 < Nout) ? col0 : (Nout - 1);  // clamped load index (values discarded)
  int colm1 = (col1 < Nout) ? col1 : (Nout - 1);

  float sc0 = rowscale ? rowscale[(long)t * M + row0] : 1.0f;
  float sc1 = rowscale ? rowscale[(long)t * M + row1] : 1.0f;

  const float* ar0 = Ab + (long)row0 * K;
  const float* ar1 = Ab + (long)row1 * K;

  v8f acc00 = {}, acc01 = {}, acc10 = {}, acc11 = {};

  int Kmain = K & ~3;
  int hk = half * 2;
  #pragma unroll 2
  for (int k = 0; k < Kmain; k += 4) {
    int ka = k + hk;
    v2f a0, a1, b0, b1;
    a0.x = ar0[ka] * sc0;  a0.y = ar0[ka + 1] * sc0;
    a1.x = ar1[ka] * sc1;  a1.y = ar1[ka + 1] * sc1;
    const float* w0 = Wb + (long)ka * Nout;
    b0.x = w0[colm0];  b0.y = w0[Nout + colm0];
    b1.x = w0[colm1];  b1.y = w0[Nout + colm1];
    acc00 = WMMA4(a0, b0, acc00);
    acc01 = WMMA4(a0, b1, acc01);
    acc10 = WMMA4(a1, b0, acc10);
    acc11 = WMMA4(a1, b1, acc11);
  }
  if (Kmain < K) {
    int ka = Kmain + hk;
    int k0 = (ka     < K) ? ka     : (K - 1);   // clamped, always-valid addresses
    int k1 = (ka + 1 < K) ? ka + 1 : (K - 1);
    float z0 = (ka     < K) ? 1.f : 0.f;        // value masks (selects, not branches)
    float z1 = (ka + 1 < K) ? 1.f : 0.f;
    v2f a0, a1, b0, b1;
    a0.x = ar0[k0] * sc0 * z0;  a0.y = ar0[k1] * sc0 * z1;
    a1.x = ar1[k0] * sc1 * z0;  a1.y = ar1[k1] * sc1 * z1;
    b0.x = Wb[(long)k0 * Nout + colm0];  b0.y = Wb[(long)k1 * Nout + colm0];
    b1.x = Wb[(long)k0 * Nout + colm1];  b1.y = Wb[(long)k1 * Nout + colm1];
    acc00 = WMMA4(a0, b0, acc00);
    acc01 = WMMA4(a0, b1, acc01);
    acc10 = WMMA4(a1, b0, acc10);
    acc11 = WMMA4(a1, b1, acc11);
  }

  float* ob = out + (long)t * M * Nout;
  #pragma unroll
  for (int q = 0; q < 4; ++q) {
    int mb = m0 + ((q >> 1) << 4);
    int cc = (q & 1) ? col1 : col0;
    if (cc >= Nout) continue;
    v8f acc = (q == 0) ? acc00 : (q == 1) ? acc01 : (q == 2) ? acc10 : acc11;
    float bv = bias ? bias[cc] : 0.f;
    #pragma unroll
    for (int i = 0; i < 8; ++i) {
      int orow = mb + i + half * 8;   // C/D layout: VGPR i -> rows i / i+8
      float v = acc[i] + bv;
      if (act == 1)      v = (v >= 0.f) ? v : SLOPE * v;
      else if (act == 2) v = (v > 0.f) ? v : 0.f;
      ob[(long)orow * Nout + cc] = v;
    }
  }
}

// ---------------- edge dot products: one wave32 per edge ----------------
__global__ void dots_kernel(const float* __restrict__ h, const int* __restrict__ src,
                            const int* __restrict__ dst, float* __restrict__ out) {
  int t    = blockIdx.y;
  int lane = threadIdx.x & 31;
  int wave = threadIdx.x >> 5;
  long e = (long)blockIdx.x * 8 + wave;
  if (e >= NEDGE) return;
  int s = src[(long)t * NEDGE + e];
  int d = dst[(long)t * NEDGE + e];
  const float* hs = h + ((long)t * NNODE + s) * NCC;
  const float* hd = h + ((long)t * NNODE + d) * NCC;
  float sum = 0.f;
  for (int f = lane; f < NCC; f += 32) sum += hs[f] * hd[f];
  for (int off = 16; off > 0; off >>= 1) sum += __shfl_xor(sum, off, 32);
  if (lane == 0) out[(long)t * NEDGE + e] = sum;
}

// ---------------- host-side orchestration ----------------
extern "C" void kernel_launch(void* const* d_in, const int* in_sizes, int n_in,
                              void* d_out, int out_size, void* d_ws, size_t ws_size,
                              hipStream_t stream) {
  const float* node = (const float*)d_in[0];
  const int* psrc = (const int*)d_in[1];
  const int* pdst = (const int*)d_in[2];
  const int* nsrc = (const int*)d_in[3];
  const int* ndst = (const int*)d_in[4];
  const float* g0[9]; for (int i = 0; i < 9; ++i) g0[i] = (const float*)d_in[5 + i];
  const float* g1[9]; for (int i = 0; i < 9; ++i) g1[i] = (const float*)d_in[14 + i];
  const float* gcn0_w = (const float*)d_in[23];
  const float* gcn0_b = (const float*)d_in[24];
  const float* gcn1_w = (const float*)d_in[25];
  const float* gcn1_b = (const float*)d_in[26];
  const float* w1 = (const float*)d_in[27];
  const float* b1 = (const float*)d_in[28];
  const float* w2 = (const float*)d_in[29];
  const float* b2 = (const float*)d_in[30];

  float* ws = (float*)d_ws;
  size_t off = 0;
  float* Ws0 = ws + off; off += (size_t)T_SNAP * FIN * HID;      // evolved gcn0 weights
  float* Ws1 = ws + off; off += (size_t)T_SNAP * HID * HID;      // evolved gcn1 weights
  float* NS  = ws + off; off += (size_t)T_SNAP * NNODE;          // src norm (out-deg^-1/2)
  float* ND  = ws + off; off += (size_t)T_SNAP * NNODE;          // dst norm (in-deg^-1/2)
  float* Qt  = ws + off; off += (size_t)HID * HID;               // GRU temps
  float* Ut  = ws + off; off += (size_t)HID * HID;
  float* Rt  = ws + off; off += (size_t)HID * HID;
  float* Ht  = ws + off; off += (size_t)HID * HID;
  float* RA  = ws + off; off += (size_t)T_SNAP * NNODE * CHH;    // agg, later MLP hidden
  float* RB  = ws + off; off += (size_t)T_SNAP * NNODE * HID;    // feats0 -> feats1 -> h
  (void)ws_size; (void)in_sizes; (void)n_in; (void)out_size;

  // degrees -> normalization factors
  {
    long n = 2L * T_SNAP * NNODE;     // NS and ND are contiguous
    fill_kernel<<<(int)((n + 255) / 256), 256, 0, stream>>>(NS, 0.f, n);
    long te = (long)T_SNAP * NEDGE;
    deg_kernel<<<(int)((te + 255) / 256), 256, 0, stream>>>(psrc, pdst, NS, ND);
    norm_kernel<<<(int)((n + 255) / 256), 256, 0, stream>>>(NS, n);
  }

  // GRU weight evolutions
  gru_kernel<<<1, 1024, 0, stream>>>(g0[0], g0[1], g0[2], g0[3], g0[4], g0[5],
                                     g0[6], g0[7], g0[8], gcn0_w, Ws0,
                                     Qt, Ut, Rt, Ht, FIN, HID);
  gru_kernel<<<1, 1024, 0, stream>>>(g1[0], g1[1], g1[2], g1[3], g1[4], g1[5],
                                     g1[6], g1[7], g1[8], gcn1_w, Ws1,
                                     Qt, Ut, Rt, Ht, HID, HID);

  dim3 sb(64, 4);
  dim3 sg((NEDGE + 3) / 4, T_SNAP);

  // ---- layer 0: aggregate (F=166) then WMMA GEMM + RReLU -> RB ----
  {
    long n = (long)T_SNAP * NNODE * FIN;
    fill_kernel<<<(int)((n + 255) / 256), 256, 0, stream>>>(RA, 0.f, n);
    scatter_kernel<<<sg, sb, 0, stream>>>(node, psrc, pdst, NS, RA, FIN);
    int tilesN = (HID + 31) / 32;
    long tiles = (long)(NNODE / 32) * tilesN;
    dim3 gg((int)((tiles + 3) / 4), T_SNAP);
    wmma_gemm_kernel<<<gg, 128, 0, stream>>>(RA, Ws0, (long)FIN * HID, gcn0_b,
                                             ND, RB, NNODE, FIN, HID, 1);
  }

  // ---- layer 1: aggregate (F=200) then WMMA GEMM + RReLU -> RB ----
  {
    long n = (long)T_SNAP * NNODE * HID;
    fill_kernel<<<(int)((n + 255) / 256), 256, 0, stream>>>(RA, 0.f, n);
    scatter_kernel<<<sg, sb, 0, stream>>>(RB, psrc, pdst, NS, RA, HID);
    int tilesN = (HID + 31) / 32;
    long tiles = (long)(NNODE / 32) * tilesN;
    dim3 gg((int)((tiles + 3) / 4), T_SNAP);
    wmma_gemm_kernel<<<gg, 128, 0, stream>>>(RA, Ws1, (long)HID * HID, gcn1_b,
                                             ND, RB, NNODE, HID, HID, 1);
  }

  // ---- MLP layer 1: RB @ w1 + b1, ReLU -> RA ----
  {
    int tilesN = (CHH + 31) / 32;
    long tiles = (long)(NNODE / 32) * tilesN;
    dim3 gg((int)((tiles + 3) / 4), T_SNAP);
    wmma_gemm_kernel<<<gg, 128, 0, stream>>>(RB, w1, 0L, b1,
                                             nullptr, RA, NNODE, HID, CHH, 2);
  }
  // ---- MLP layer 2: RA @ w2 + b2 -> RB (final node embeddings h) ----
  {
    int tilesN = (NCC + 31) / 32;
    long tiles = (long)(NNODE / 32) * tilesN;
    dim3 gg((int)((tiles + 3) / 4), T_SNAP);
    wmma_gemm_kernel<<<gg, 128, 0, stream>>>(RA, w2, 0L, b2,
                                             nullptr, RB, NNODE, CHH, NCC, 0);
  }

  // ---- edge dot products ----
  float* outp = (float*)d_out;
  dim3 dg((NEDGE + 7) / 8, T_SNAP);
  dots_kernel<<<dg, 256, 0, stream>>>(RB, psrc, pdst, outp);
  dots_kernel<<<dg, 256, 0, stream>>>(RB, nsrc, ndst, outp + (long)T_SNAP * NEDGE);
}